// MultiHeadAdditiveAttention_25374666785353
// MI455X (gfx1250) — compile-verified
//
#include <hip/hip_runtime.h>
#include <hip/hip_bf16.h>

// ---- problem constants (from reference) ----
#define BB   16
#define SS   64
#define HH   16
#define DD   16
#define EE   256      // H*D
#define HID  128
#define MQ   (BB*SS)          // 1024 rows for QKV / FC GEMMs
#define MPRE (BB*SS*HH)       // 16384 rows for pre_q/pre_k GEMM

typedef float v2f __attribute__((ext_vector_type(2)));
typedef float v8f __attribute__((ext_vector_type(8)));

__device__ __forceinline__ float fast_tanh(float x) {
#if __has_builtin(__builtin_amdgcn_tanhf)
    return __builtin_amdgcn_tanhf(x);
#else
    return tanhf(x);
#endif
}

__device__ __forceinline__ v2f ld2(const float* p) {
    return *reinterpret_cast<const v2f*>(p);
}

// ---------------------------------------------------------------------------
// Generic fp32 GEMM on the matrix pipe:
//   C[m, n] = sum_k A[m*lda + k] * W[n*ldw + wOff + k]   (+ bias[n])
// One 16x16 output tile per wave, V_WMMA_F32_16X16X4_F32, K unrolled by 16.
// Requires: M%16==0, N%16==0, K%16==0, grid sized so every wave has a tile.
// ---------------------------------------------------------------------------
#define GEMM_WAVES 8

__global__ __launch_bounds__(256) void gemm_bias_wmma_f32(
    const float* __restrict__ A, int lda,
    const float* __restrict__ W, int ldw, int wOff,
    const float* __restrict__ bias,
    float* __restrict__ C, int ldc,
    int tilesN, int K)
{
    const int lane = threadIdx.x;            // 0..31
    const int wave = threadIdx.y;            // 0..7
    const int gw   = blockIdx.x * GEMM_WAVES + wave;
    const int tm   = gw / tilesN;
    const int tn   = gw - tm * tilesN;
    const int m0   = tm * 16;
    const int n0   = tn * 16;
    const int half = lane >> 4;              // 0 or 1
    const int sub  = lane & 15;
    const int h2   = half * 2;

    const float* aRow = A + (size_t)(m0 + sub) * lda;
    const float* wRow = W + (size_t)(n0 + sub) * ldw + wOff;

    v8f c = {};
    for (int k0 = 0; k0 < K; k0 += 16) {
        // per-lane pairs for 4 consecutive K=4 WMMA steps
        v2f a0 = ld2(aRow + k0 +  0 + h2);
        v2f a1 = ld2(aRow + k0 +  4 + h2);
        v2f a2 = ld2(aRow + k0 +  8 + h2);
        v2f a3 = ld2(aRow + k0 + 12 + h2);
        v2f b0 = ld2(wRow + k0 +  0 + h2);
        v2f b1 = ld2(wRow + k0 +  4 + h2);
        v2f b2 = ld2(wRow + k0 +  8 + h2);
        v2f b3 = ld2(wRow + k0 + 12 + h2);
        c = __builtin_amdgcn_wmma_f32_16x16x4_f32(false, a0, false, b0, (short)0, c, false, false);
        c = __builtin_amdgcn_wmma_f32_16x16x4_f32(false, a1, false, b1, (short)0, c, false, false);
        c = __builtin_amdgcn_wmma_f32_16x16x4_f32(false, a2, false, b2, (short)0, c, false, false);
        c = __builtin_amdgcn_wmma_f32_16x16x4_f32(false, a3, false, b3, (short)0, c, false, false);
    }

    const int   col  = n0 + sub;
    const int   row0 = m0 + half * 8;
    const float bb   = bias ? bias[col] : 0.0f;
#pragma unroll
    for (int r = 0; r < 8; ++r)
        C[(size_t)(row0 + r) * ldc + col] = c[r] + bb;
}

// ---------------------------------------------------------------------------
// Fused scores -> softmax -> ctx.  One block per (b, i) query position.
//   scores[j,h] = sum_c tanh(pre_q[b,i,h,c] + pre_k[b,j,h,c]) * attn_v[c]
//   w = softmax_j(scores)
//   ctx[b,i, j*D+d] = sum_h w[j,h] * v[b,i,h,d]
// 256 threads = 8 waves; each wave owns one (j,h) pair at a time with the
// 128-long c-axis spread 4-per-lane (fully coalesced float4 pre_k streams),
// then a 5-step shuffle reduction.
// ---------------------------------------------------------------------------
__global__ __launch_bounds__(256) void scores_softmax_ctx(
    const float* __restrict__ pre_q,   // [B*S, H*HID]
    const float* __restrict__ pre_k,   // [B*S, H*HID]
    const float* __restrict__ attn_v,  // [HID]
    const float* __restrict__ v,       // [B*S, H*D]
    float* __restrict__ ctx)           // [B*S, S*D]
{
    __shared__ float pq[HH * HID];     // 8 KB
    __shared__ float av[HID];
    __shared__ float sc[SS * HH];      // scores then softmax weights, 4 KB
    __shared__ float vloc[HH * DD];    // 1 KB

    const int    tid = threadIdx.x;
    const size_t bi  = blockIdx.x;           // b*S + i
    const size_t bat = bi / SS;              // b

    const float* pqg = pre_q + bi * (HH * HID);
    for (int t = tid; t < HH * HID; t += 256) pq[t] = pqg[t];
    for (int t = tid; t < HID;      t += 256) av[t] = attn_v[t];
    const float* vg = v + bi * (HH * DD);
    for (int t = tid; t < HH * DD;  t += 256) vloc[t] = vg[t];
    __syncthreads();

    const int lane = tid & 31;
    const int wv   = tid >> 5;
    const int c0   = lane * 4;
    const float a0 = av[c0], a1 = av[c0 + 1], a2 = av[c0 + 2], a3 = av[c0 + 3];
    const float* pkb = pre_k + bat * (size_t)(SS * HH * HID);

    for (int p = wv; p < SS * HH; p += 8) {
        const int j = p >> 4;
        const int h = p & 15;
        const float* pk  = pkb + ((size_t)j * HH + h) * HID + c0;  // coalesced across wave
        const float* pqh = pq + h * HID + c0;
        float acc = fast_tanh(pqh[0] + pk[0]) * a0
                  + fast_tanh(pqh[1] + pk[1]) * a1
                  + fast_tanh(pqh[2] + pk[2]) * a2
                  + fast_tanh(pqh[3] + pk[3]) * a3;
#pragma unroll
        for (int off = 16; off > 0; off >>= 1)
            acc += __shfl_xor(acc, off, 32);
        if (lane == 0) sc[j * HH + h] = acc;
    }
    __syncthreads();

    // softmax over j (key axis) independently per head h
    if (tid < HH) {
        const int h = tid;
        float m = -3.0e38f;
        for (int j = 0; j < SS; ++j) m = fmaxf(m, sc[j * HH + h]);
        float s = 0.0f;
        for (int j = 0; j < SS; ++j) {
            float e = __expf(sc[j * HH + h] - m);
            sc[j * HH + h] = e;
            s += e;
        }
        const float inv = 1.0f / s;
        for (int j = 0; j < SS; ++j) sc[j * HH + h] *= inv;
    }
    __syncthreads();

    // ctx[b,i,j,d] = sum_h w[j,h] * v[b,i,h,d]
    float* cg = ctx + bi * (size_t)(SS * DD);
    for (int idx = tid; idx < SS * DD; idx += 256) {
        const int j = idx >> 4;
        const int d = idx & 15;
        float acc = 0.0f;
#pragma unroll
        for (int h = 0; h < HH; ++h)
            acc = fmaf(sc[j * HH + h], vloc[h * DD + d], acc);
        cg[idx] = acc;            // coalesced
    }
}

// ---------------------------------------------------------------------------
extern "C" void kernel_launch(void* const* d_in, const int* in_sizes, int n_in,
                              void* d_out, int out_size, void* d_ws, size_t ws_size,
                              hipStream_t stream) {
    (void)in_sizes; (void)n_in; (void)out_size; (void)ws_size;
    const float* x      = (const float*)d_in[0];
    const float* Wq     = (const float*)d_in[1];
    const float* bq     = (const float*)d_in[2];
    const float* Wk     = (const float*)d_in[3];
    const float* bk     = (const float*)d_in[4];
    const float* Wv     = (const float*)d_in[5];
    const float* bv     = (const float*)d_in[6];
    const float* Wa     = (const float*)d_in[7];   // [HID, 2D]
    const float* ba     = (const float*)d_in[8];
    const float* attn_v = (const float*)d_in[9];
    const float* Wfc    = (const float*)d_in[10];  // [E, S*D]
    const float* bfc    = (const float*)d_in[11];
    float* out = (float*)d_out;

    // workspace layout (floats)
    float* q     = (float*)d_ws;                    // [1024, 256]
    float* k     = q     + (size_t)MQ * EE;         // [1024, 256]
    float* v     = k     + (size_t)MQ * EE;         // [1024, 256]
    float* pre_q = v     + (size_t)MQ * EE;         // [16384, 128]
    float* pre_k = pre_q + (size_t)MPRE * HID;      // [16384, 128]
    float* ctx   = pre_k + (size_t)MPRE * HID;      // [1024, 1024]

    const dim3 blk(32, GEMM_WAVES);

    // Q/K/V projections: M=1024, N=256, K=256 -> 1024 tiles -> 128 blocks
    gemm_bias_wmma_f32<<<128, blk, 0, stream>>>(x, EE, Wq, EE, 0, bq, q, EE, EE / 16, EE);
    gemm_bias_wmma_f32<<<128, blk, 0, stream>>>(x, EE, Wk, EE, 0, bk, k, EE, EE / 16, EE);
    gemm_bias_wmma_f32<<<128, blk, 0, stream>>>(x, EE, Wv, EE, 0, bv, v, EE, EE / 16, EE);

    // pre_q / pre_k: M=16384, N=128, K=16 -> 8192 tiles -> 1024 blocks
    // pre_q uses Wa[:, :D] (+ba), pre_k uses Wa[:, D:] (no bias)
    gemm_bias_wmma_f32<<<1024, blk, 0, stream>>>(q, DD, Wa, 2 * DD, 0,  ba,      pre_q, HID, HID / 16, DD);
    gemm_bias_wmma_f32<<<1024, blk, 0, stream>>>(k, DD, Wa, 2 * DD, DD, nullptr, pre_k, HID, HID / 16, DD);

    // fused scores -> softmax -> ctx : one block per (b, i)
    scores_softmax_ctx<<<BB * SS, 256, 0, stream>>>(pre_q, pre_k, attn_v, v, ctx);

    // final FC: M=1024, N=256, K=1024 -> 1024 tiles -> 128 blocks
    gemm_bias_wmma_f32<<<128, blk, 0, stream>>>(ctx, SS * DD, Wfc, SS * DD, 0, bfc, out, EE, EE / 16, SS * DD);
}